// QAELayer_46188078301220
// MI455X (gfx1250) — compile-verified
//
#include <hip/hip_runtime.h>
#include <math.h>

// QAE attention layer for MI455X (gfx1250, wave32).
// Bandwidth-bound: 201MB read (pool) + 201MB read (scale, mostly L2-hit: 192MB L2)
// + 201MB NT write  =>  ~17-26us floor @ 23.3 TB/s. VQC math is negligible; its
// fc GEMM (64x12x4) is done with v_wmma_f32_16x16x4_f32.

typedef __attribute__((ext_vector_type(2))) float v2f;
typedef __attribute__((ext_vector_type(4))) float v4f;
typedef __attribute__((ext_vector_type(8))) float v8f;

#define HWQ   16384        // 65536 elements / 4 per plane, as float4
#define NPLN  768          // B*C = 64*12 planes

// ---------------------------------------------------------------- pool ------
// One block per (b,c) plane: mean over 256x256. Default (RT) loads keep x
// resident in the 192MB L2 for the scale pass.
__global__ __launch_bounds__(256) void pool_kernel(const float* __restrict__ x,
                                                   float* __restrict__ y) {
  __shared__ float red[256];
  const int p   = blockIdx.x;
  const int tid = threadIdx.x;
  const v4f* xp = (const v4f*)(x + (size_t)p * 65536);
  float sum = 0.0f;
  for (int j = tid; j < HWQ; j += 256) {
    v4f v = xp[j];
    sum += v.x + v.y + v.z + v.w;
  }
  red[tid] = sum;
  __syncthreads();
  for (int s = 128; s > 0; s >>= 1) {
    if (tid < s) red[tid] += red[tid + s];
    __syncthreads();
  }
  if (tid == 0) y[p] = red[0] * (1.0f / 65536.0f);
}

// ----------------------------------------------------------------- vqc ------
// 1 block, 64 threads (2 waves). Each thread simulates one batch element's
// 4-qubit statevector entirely in registers, then both waves run the
// 64x12 = [64x4]@[4x16] fc GEMM with V_WMMA_F32_16X16X4_F32 (2 tiles/wave).
__global__ __launch_bounds__(64) void vqc_kernel(const float* __restrict__ y,
                                                 const float* __restrict__ wt,
                                                 const float* __restrict__ fw,
                                                 const float* __restrict__ fb,
                                                 float* __restrict__ att) {
  __shared__ float zl[64 * 4];   // z-expectations, [b][q]
  __shared__ float fwl[48];      // fc_w [12][4]
  __shared__ float fbl[12];      // fc_b
  const int tid = threadIdx.x;
  if (tid < 48) fwl[tid] = fw[tid];
  if (tid < 12) fbl[tid] = fb[tid];

  // ---- per-thread statevector simulation (b = tid) ----
  float sr[16], si[16];
  {
    // Encoding: per-qubit Rz(a0)Ry(a1)Rz(a2) on |+>; state stays a product
    // state, so build 4 complex 2-vectors then take the tensor product.
    float vr[4][2], vi[4][2];
    const float h = 0.70710678118654752f;   // 1/sqrt(2)
#pragma unroll
    for (int w = 0; w < 4; ++w) {
      const float a0 = y[tid * 12 + w * 3 + 0];
      const float a1 = y[tid * 12 + w * 3 + 1];
      const float a2 = y[tid * 12 + w * 3 + 2];
      float c0, s0; __sincosf(0.5f * a0, &s0, &c0);
      float v0r = h * c0, v0i = -h * s0;           // h * e^{-i a0/2}
      float v1r = h * c0, v1i =  h * s0;           // h * e^{+i a0/2}
      float c1, s1; __sincosf(0.5f * a1, &s1, &c1);
      float w0r = c1 * v0r - s1 * v1r, w0i = c1 * v0i - s1 * v1i;
      float w1r = s1 * v0r + c1 * v1r, w1i = s1 * v0i + c1 * v1i;
      float c2, s2; __sincosf(0.5f * a2, &s2, &c2);
      vr[w][0] = w0r * c2 + w0i * s2;  vi[w][0] = w0i * c2 - w0r * s2;
      vr[w][1] = w1r * c2 - w1i * s2;  vi[w][1] = w1i * c2 + w1r * s2;
    }
#pragma unroll
    for (int i = 0; i < 16; ++i) {
      float pr = vr[0][(i >> 3) & 1], pi = vi[0][(i >> 3) & 1];
#pragma unroll
      for (int w = 1; w < 4; ++w) {
        const int bit = (i >> (3 - w)) & 1;
        const float qr = vr[w][bit], qi = vi[w][bit];
        const float nr = pr * qr - pi * qi;
        const float ni = pr * qi + pi * qr;
        pr = nr; pi = ni;
      }
      sr[i] = pr; si[i] = pi;
    }
  }

  // ---- CNOT chain (0,1)(1,2)(2,3)(3,0); wire w -> bit (3-w) ----
  {
    const int cn[4][2] = {{8, 4}, {4, 2}, {2, 1}, {1, 8}};
#pragma unroll
    for (int g = 0; g < 4; ++g) {
      const int cb = cn[g][0], tb = cn[g][1];
#pragma unroll
      for (int i = 0; i < 16; ++i) {
        if ((i & cb) && !(i & tb)) {
          const int j = i | tb;
          float t;
          t = sr[i]; sr[i] = sr[j]; sr[j] = t;
          t = si[i]; si[i] = si[j]; si[j] = t;
        }
      }
    }
  }

  // ---- U3 layer (weights shared across batch) ----
#pragma unroll
  for (int w = 0; w < 4; ++w) {
    const float th = wt[w * 3 + 0], ph = wt[w * 3 + 1], la = wt[w * 3 + 2];
    float ct, st;  __sincosf(0.5f * th, &st, &ct);
    float cl, sl;  __sincosf(la, &sl, &cl);
    float cp, sp;  __sincosf(ph, &sp, &cp);
    float cq, sq;  __sincosf(ph + la, &sq, &cq);
    const float u00r = ct,        u00i = 0.0f;
    const float u01r = -cl * st,  u01i = -sl * st;
    const float u10r =  cp * st,  u10i =  sp * st;
    const float u11r =  cq * ct,  u11i =  sq * ct;
    const int wb = 1 << (3 - w);
#pragma unroll
    for (int i = 0; i < 16; ++i) {
      if (!(i & wb)) {
        const int j = i | wb;
        const float a0r = sr[i], a0i = si[i], a1r = sr[j], a1i = si[j];
        sr[i] = u00r * a0r - u00i * a0i + u01r * a1r - u01i * a1i;
        si[i] = u00r * a0i + u00i * a0r + u01r * a1i + u01i * a1r;
        sr[j] = u10r * a0r - u10i * a0i + u11r * a1r - u11i * a1i;
        si[j] = u10r * a0i + u10i * a0r + u11r * a1i + u11i * a1r;
      }
    }
  }

  // ---- Z expectations ----
#pragma unroll
  for (int w = 0; w < 4; ++w) {
    const int wb = 1 << (3 - w);
    float zv = 0.0f;
#pragma unroll
    for (int i = 0; i < 16; ++i) {
      const float pp = sr[i] * sr[i] + si[i] * si[i];
      zv += (i & wb) ? -pp : pp;
    }
    zl[tid * 4 + w] = zv;
  }
  __syncthreads();

  // ---- fc via WMMA: P[64,16] = Z[64,4] @ B[4,16], B[k][n] = fc_w[n*4+k] ----
  const int lane = tid & 31;
  const int wave = tid >> 5;
  const int n    = lane & 15;
  const int kb   = (lane >= 16) ? 2 : 0;      // K-half per ISA A/B layouts
  v2f bm;
  bm.x = (n < 12) ? fwl[n * 4 + kb]     : 0.0f;   // B: K=kb,   N=n
  bm.y = (n < 12) ? fwl[n * 4 + kb + 1] : 0.0f;   // B: K=kb+1, N=n
  const float bias = (n < 12) ? fbl[n] : 0.0f;

#pragma unroll
  for (int t = 0; t < 2; ++t) {
    const int tile = wave * 2 + t;            // row tile: rows [tile*16, +16)
    const int row  = tile * 16 + n;           // A: lanes 0-15/16-31 both M=0..15
    v2f a;
    a.x = zl[row * 4 + kb];
    a.y = zl[row * 4 + kb + 1];
    v8f acc = {};
    acc = __builtin_amdgcn_wmma_f32_16x16x4_f32(
        /*neg_a=*/false, a, /*neg_b=*/false, bm,
        /*c_mod=*/(short)0, acc, /*reuse_a=*/false, /*reuse_b=*/false);
    // D layout: VGPR r -> (M = r + 8*(lane>=16), N = lane&15)
#pragma unroll
    for (int r = 0; r < 8; ++r) {
      const int m = tile * 16 + r + ((lane >= 16) ? 8 : 0);
      if (n < 12) {
        const float pre = acc[r] + bias;
        att[m * 12 + n] = 1.0f / (1.0f + __expf(-pre));
      }
    }
  }
}

// --------------------------------------------------------------- scale ------
// One block per plane. x loads hit L2 (warmed by pool pass); output stores are
// non-temporal so 201MB of writes don't evict x mid-pass.
__global__ __launch_bounds__(256) void scale_kernel(const float* __restrict__ x,
                                                    const float* __restrict__ att,
                                                    float* __restrict__ out) {
  const int p   = blockIdx.x;
  const int tid = threadIdx.x;
  const float a = att[p];
  const v4f* xp = (const v4f*)(x   + (size_t)p * 65536);
  v4f*       op = (v4f*)      (out + (size_t)p * 65536);
  for (int j = tid; j < HWQ; j += 256) {
    v4f v = xp[j];
    v *= a;
    __builtin_nontemporal_store(v, &op[j]);
  }
}

// ------------------------------------------------------------- launcher -----
extern "C" void kernel_launch(void* const* d_in, const int* in_sizes, int n_in,
                              void* d_out, int out_size, void* d_ws, size_t ws_size,
                              hipStream_t stream) {
  (void)in_sizes; (void)n_in; (void)out_size; (void)ws_size;
  const float* x  = (const float*)d_in[0];   // [64,12,256,256]
  const float* wt = (const float*)d_in[1];   // [1,4,3]
  const float* fw = (const float*)d_in[2];   // [12,4]
  const float* fb = (const float*)d_in[3];   // [12]
  float* out = (float*)d_out;                // [64,12,256,256]
  float* y   = (float*)d_ws;                 // [768] pooled means
  float* att = y + NPLN;                     // [768] attention

  pool_kernel <<<NPLN, 256, 0, stream>>>(x, y);
  vqc_kernel  <<<1,    64,  0, stream>>>(y, wt, fw, fb, att);
  scale_kernel<<<NPLN, 256, 0, stream>>>(x, att, out);
}